// Attn_32332513804638
// MI455X (gfx1250) — compile-verified
//
#include <hip/hip_runtime.h>

typedef float v2f __attribute__((ext_vector_type(2)));
typedef float v8f __attribute__((ext_vector_type(8)));
typedef __attribute__((address_space(3))) float lds_f;

#define N_    16
#define M_    100
#define LOC_  10000
#define EMB_  128
#define D_    4
#define LTILES (LOC_ / 16)   /* 625 l-tiles */
#define DROW  66             /* LDS row stride (dwords): bank-conflict-free, 8B aligned rows */
#define BUFSZ (16 * DROW)    /* one delta chunk buffer, in floats */

__global__ __launch_bounds__(32)
void attn_wmma_kernel(const float* __restrict__ self_attn,   // [N,M,EMB]
                      const float* __restrict__ self_delta,  // [N,M,LOC,D]
                      const float* __restrict__ emb_table,   // [LOC+1,EMB]
                      const float* __restrict__ value_w,     // [M]
                      float* __restrict__ out)               // [N,LOC]
{
    __shared__ float lds[2 * BUFSZ];   // double-buffered staged delta chunks

    const int lane = threadIdx.x;      // 0..31 (wave32)
    const int col  = lane & 15;
    const int hi   = lane >> 4;

    const int blk = blockIdx.x;
    const int n   = blk / LTILES;
    const int l0  = (blk % LTILES) << 4;

    const float* attn_n  = self_attn  + (size_t)n * M_ * EMB_;
    const float* delta_n = self_delta + (size_t)n * M_ * (size_t)LOC_ * D_;

    // ---- async stage of one delta chunk into LDS buffer `bufsel` ----
    // Row mm = self_delta[n][m0+mm][l0..l0+15][0..3]: 64 contiguous floats (256B).
    // One wave-wide global_load_async_to_lds_b64 per row: HBM -> LDS directly,
    // non-temporal (zero-reuse 256MB stream), tracked on ASYNCcnt so it overlaps
    // the WMMA loop.
    auto stage = [&](int chunk, int bufsel) {
#pragma unroll
        for (int mm = 0; mm < 16; ++mm) {
            const int m = min(chunk * 16 + mm, M_ - 1);   // clamp; invalid cols killed by w=0
            const float* g = delta_n + ((size_t)m * LOC_ + l0) * D_ + 2 * lane;
            lds_f* l = (lds_f*)&lds[bufsel * BUFSZ + mm * DROW + 2 * lane];
            asm volatile("global_load_async_to_lds_b64 %0, %1, off th:TH_LOAD_NT"
                         :: "v"(l), "v"(g) : "memory");
        }
    };

    stage(0, 0);   // prefetch chunk 0 before doing anything else

    // ---- preload A tile (emb_cand rows l0..l0+15, all 128 K) into registers ----
    // A layout for V_WMMA_F32_16X16X4_F32:
    //   lanes 0-15 hold matrix rows (M = l); VGPR0: K = k0+2*hi, VGPR1: K = k0+2*hi+1
    const float* embrow = emb_table + (size_t)(1 + l0 + col) * EMB_ + 2 * hi;
    v2f a[32];
#pragma unroll
    for (int k = 0; k < 32; ++k)
        a[k] = *(const v2f*)(embrow + 4 * k);

    float oacc[8];
#pragma unroll
    for (int i = 0; i < 8; ++i) oacc[i] = 0.0f;

    for (int chunk = 0; chunk < 7; ++chunk) {
        const int m0     = chunk << 4;
        const int bufsel = chunk & 1;

        if (chunk < 6)
            stage(chunk + 1, bufsel ^ 1);   // overlap next chunk's HBM stream with compute

        // ---- batch-load the chunk's B operands into registers first ----
        // (explicit array forces a deep load clause -> partial loadcnt waits,
        //  instead of a serialized load->wmma->load->wmma chain)
        const int   mB  = m0 + col;
        const bool  mOK = (mB < M_);
        const float wv  = mOK ? value_w[mB] : 0.0f;
        const float* arow = attn_n + (size_t)(mOK ? mB : (M_ - 1)) * EMB_ + 2 * hi;

        v2f b[32];
#pragma unroll
        for (int k = 0; k < 32; ++k)
            b[k] = *(const v2f*)(arow + 4 * k);

        // ---- S = emb_tile(16x128) x attn_chunk^T(128x16): 32 K-steps of WMMA ----
        v8f acc = {};
#pragma unroll
        for (int k = 0; k < 32; ++k)
            acc = __builtin_amdgcn_wmma_f32_16x16x4_f32(
                false, a[k], false, b[k], (short)0, acc, false, false);

        // Async loads retire in order: <=16 outstanding means this chunk's 16 are done.
        if (chunk < 6)
            asm volatile("s_wait_asynccnt 0x10" ::: "memory");
        else
            asm volatile("s_wait_asynccnt 0x0" ::: "memory");

        // ---- combine: oacc[i] += S[l,m] * (sum_d delta[n,m,l,d]) * w[m] ----
        // C/D layout: VGPR i -> lanes 0-15: l = i (m = col); lanes 16-31: l = 8 + i
#pragma unroll
        for (int i = 0; i < 8; ++i) {
            const int lrow = i + 8 * hi;
            const float* p = &lds[bufsel * BUFSZ + col * DROW + lrow * 4];
            v2f d01 = *(const v2f*)(p);
            v2f d23 = *(const v2f*)(p + 2);
            float ds = (d01.x + d01.y) + (d23.x + d23.y);
            oacc[i] += acc[i] * ds * wv;
        }
    }

    // ---- reduce the m-sum across the 16 lanes of each half-wave, write out ----
#pragma unroll
    for (int i = 0; i < 8; ++i) {
        float v = oacc[i];
#pragma unroll
        for (int off = 8; off >= 1; off >>= 1)
            v += __shfl_xor(v, off, 32);
        if (col == 0)
            out[(size_t)n * LOC_ + l0 + i + 8 * hi] = v;
    }
}

extern "C" void kernel_launch(void* const* d_in, const int* in_sizes, int n_in,
                              void* d_out, int out_size, void* d_ws, size_t ws_size,
                              hipStream_t stream) {
    const float* self_attn  = (const float*)d_in[0];   // (N, M, EMB) f32
    const float* self_delta = (const float*)d_in[1];   // (N, M, LOC, D) f32
    // d_in[2] = traj_len (int) — unused by the reference computation
    const float* emb_table  = (const float*)d_in[3];   // (LOC+1, EMB) f32
    const float* value_w    = (const float*)d_in[4];   // (M,) f32
    float* out = (float*)d_out;                        // (N, LOC) f32

    attn_wmma_kernel<<<dim3(N_ * LTILES), dim3(32), 0, stream>>>(
        self_attn, self_delta, emb_table, value_w, out);
}